// CTCLoss_73856257622407
// MI455X (gfx1250) — compile-verified
//
#include <hip/hip_runtime.h>
#include <stdint.h>

// ---------------------------------------------------------------------------
// CTC forward loss for fixed shapes T=512, B=1024, C=63, S=25 (L=51).
// One wave32 per batch element; lane k owns extended states 2k and 2k+1.
// log_probs rows are streamed 8-deep into LDS with the CDNA5 async copy
// engine (global_load_async_to_lds_b64 / s_wait_asynccnt); the sequential
// DP consumes them via LDS gathers + ds_bpermute for the cross-state term.
// ---------------------------------------------------------------------------

#define NEGV (-1e30f)

__device__ __forceinline__ void async_row_load(unsigned lds_addr, unsigned long long gaddr) {
  // CDNA5 async global -> LDS copy, 8 bytes per lane, tracked by ASYNCcnt.
  asm volatile("global_load_async_to_lds_b64 %0, %1, off"
               :: "v"(lds_addr), "v"(gaddr)
               : "memory");
}

template <int N>
__device__ __forceinline__ void wait_async() {
  asm volatile("s_wait_asynccnt %0" :: "n"(N) : "memory");
}

__global__ __launch_bounds__(256) void ctc_alpha_kernel(
    const float* __restrict__ lp,      // [T,B,C] log-probs
    const int* __restrict__ targets,   // [B,S]
    const int* __restrict__ tlen,      // [B]
    float* __restrict__ ws) {          // [B] per-batch nll/len
  constexpr int T = 512, B = 1024, C = 63, S = 25;
  constexpr unsigned STRIDE = (unsigned)B * C * 4;  // bytes between timesteps

  // 8 waves/block, each wave gets 8 ring slots of 64 floats (256 B).
  __shared__ float smem[8 * 8 * 64];

  const int w    = threadIdx.x >> 5;
  const int lane = threadIdx.x & 31;
  const int b    = (blockIdx.x << 3) + w;

  // Per-lane extended-target info. Lane k: odd state label = targets[b][k].
  int tg = 62, tgp = 62;                 // dummy in-row index for lanes >= S
  if (lane < S)              tg  = targets[b * S + lane];
  if (lane >= 1 && lane <= S) tgp = targets[b * S + lane - 1];
  const bool skip_o = (lane == 0) || (tg != tgp);  // labels are >=1, never blank
  const int len = tlen[b];

  float* sw = smem + (w << 9);                       // this wave's 512 floats
  const unsigned lds_w = (unsigned)(uintptr_t)sw;    // LDS byte offset (low 32 bits)
  unsigned lofs = (unsigned)lane << 3;
  if (lofs > 244u) lofs = 244u;                      // clamp: row is 252 bytes
  const unsigned long long grow0 =
      (unsigned long long)(uintptr_t)lp + (unsigned long long)b * (C * 4);

  // Prologue: stage rows t = 0..7 into the ring.
#pragma unroll
  for (int t = 0; t < 8; ++t)
    async_row_load(lds_w + ((unsigned)t << 8) + lofs,
                   grow0 + (unsigned long long)t * STRIDE + lofs);

  // ---- t = 0 init: only states 0 (blank) and 1 (first label) reachable ----
  wait_async<7>();
  float a_e, a_o;
  {
    const float* r0 = sw;
    const float lpb = r0[0];
    const float lpl = r0[tg];
    a_e = (lane == 0) ? lpb : NEGV;
    a_o = (lane == 0) ? lpl : NEGV;
  }
  async_row_load(lds_w + lofs, grow0 + 8ull * STRIDE + lofs);  // t=8 -> slot 0

  // One DP step: consumes the staged row r, updates (a_e, a_o).
  auto step = [&](const float* r) {
    const float lpb = r[0];
    const float lpl = r[tg];
    float ao_prev = __int_as_float(__builtin_amdgcn_ds_bpermute(
        ((lane - 1) & 31) << 2, __float_as_int(a_o)));
    ao_prev = (lane == 0) ? NEGV : ao_prev;
    // even state 2k: logsumexp(alpha[2k], alpha[2k-1]) + lp[blank]
    const float m2 = fmaxf(a_e, ao_prev);
    const float ne = m2 + __logf(__expf(a_e - m2) + __expf(ao_prev - m2)) + lpb;
    // odd state 2k+1: logsumexp(alpha[2k+1], alpha[2k], skip? alpha[2k-1]) + lp[tgt_k]
    const float c3 = skip_o ? ao_prev : NEGV;
    const float m3 = fmaxf(fmaxf(a_o, a_e), c3);
    const float s3 = __expf(a_o - m3) + __expf(a_e - m3) +
                     (skip_o ? __expf(ao_prev - m3) : 0.0f);
    const float no = m3 + __logf(s3) + lpl;
    a_e = ne;
    a_o = no;
  };

  // ---- steady state: compute t, prefetch t+8 ----
  for (int t = 1; t < T - 8; ++t) {
    wait_async<7>();  // in-order completion => row t has landed
    step(sw + ((t & 7) << 6));
    async_row_load(lds_w + (((unsigned)t & 7u) << 8) + lofs,
                   grow0 + (unsigned long long)(t + 8) * STRIDE + lofs);
  }

  // ---- drain: remaining 8 rows are all in flight or landed ----
  wait_async<0>();
  for (int t = T - 8; t < T; ++t) step(sw + ((t & 7) << 6));

  // ---- finalize: ll = logaddexp(alpha[2*len-1], alpha[2*len]) ----
  if (lane < 26) {
    sw[lane]      = a_e;  // state 2k
    sw[32 + lane] = a_o;  // state 2k+1
  }
  if (lane == 0) {
    const float ahi = sw[len];            // alpha[2*len]
    const float alo = sw[32 + len - 1];   // alpha[2*len-1]
    const float m  = fmaxf(ahi, alo);
    const float ll = m + __logf(__expf(ahi - m) + __expf(alo - m));
    float nll = -ll;
    if (nll > 0.5e30f) nll = 0.0f;        // zero_infinity
    const int dl = (len > 1) ? len : 1;
    ws[b] = nll / (float)dl;
  }
}

__global__ __launch_bounds__(256) void ctc_reduce_kernel(
    const float* __restrict__ ws, float* __restrict__ out) {
  __shared__ float s[256];
  const int tid = threadIdx.x;
  s[tid] = ws[tid] + ws[tid + 256] + ws[tid + 512] + ws[tid + 768];
  __syncthreads();
#pragma unroll
  for (int st = 128; st > 0; st >>= 1) {
    if (tid < st) s[tid] += s[tid + st];
    __syncthreads();
  }
  if (tid == 0) out[0] = s[0] * (1.0f / 1024.0f);
}

extern "C" void kernel_launch(void* const* d_in, const int* in_sizes, int n_in,
                              void* d_out, int out_size, void* d_ws, size_t ws_size,
                              hipStream_t stream) {
  const float* lp      = (const float*)d_in[0];  // [512,1024,63] f32
  const int*   targets = (const int*)d_in[1];    // [1024,25] i32
  const int*   tlen    = (const int*)d_in[2];    // [1024] i32
  float* ws  = (float*)d_ws;                     // 1024 floats of scratch
  float* out = (float*)d_out;                    // scalar f32

  // 1024 batch elements, 8 waves (8 batch rows) per 256-thread block.
  ctc_alpha_kernel<<<128, 256, 0, stream>>>(lp, targets, tlen, ws);
  ctc_reduce_kernel<<<1, 256, 0, stream>>>(ws, out);
}